// VectorQuantizer_L2norm_38663295599459
// MI455X (gfx1250) — compile-verified
//
#include <hip/hip_runtime.h>

// ---------------------------------------------------------------------------
// VQ quantizer for MI455X (gfx1250, wave32).
// - GEMM via V_WMMA_F32_16X16X4_F32 (fp32 matrix path), flash-softmax 2-pass.
// - B tiles staged LDS-side by the Tensor Data Mover (TENSOR_LOAD_TO_LDS),
//   double-buffered, synchronized with s_wait_tensorcnt. Manual fallback kept.
// - LDS fragments in float2 layout -> single ds_load_b64 per WMMA operand.
// ---------------------------------------------------------------------------

#define N_ROWS  8192
#define K_CODES 8192
#define DIM     256
#define BLOCK_ROWS 64    // rows per workgroup in GEMM passes (4 waves x 16)
#define KT 16            // codes per k-tile
#define B_ROW_DW 258     // 256 dwords + 2 pad dwords (TDM pad) -> no 16-way bank conflict
#define ND_TOT (N_ROWS * DIM)

typedef float v2f __attribute__((ext_vector_type(2)));
typedef float v8f __attribute__((ext_vector_type(8)));
typedef unsigned int v4u __attribute__((ext_vector_type(4)));
typedef int v4i __attribute__((ext_vector_type(4)));
typedef int v8i __attribute__((ext_vector_type(8)));

#if defined(__gfx1250__) && __has_builtin(__builtin_amdgcn_tensor_load_to_lds)
#define USE_TDM 1
#else
#define USE_TDM 0
#endif

__device__ __forceinline__ float grp16_max(float v) {
#pragma unroll
  for (int off = 1; off < 16; off <<= 1) v = fmaxf(v, __shfl_xor(v, off, 16));
  return v;
}
__device__ __forceinline__ float grp16_sum(float v) {
#pragma unroll
  for (int off = 1; off < 16; off <<= 1) v += __shfl_xor(v, off, 16);
  return v;
}

#if USE_TDM
// DMA 4 rows x 256 floats of en (row stride 256 floats) into LDS at
// lds_byte_addr, inserting 2 pad DWORDs after every 256 DWORDs (row stride
// becomes B_ROW_DW). One descriptor per wave -> 4 descriptors cover a KT tile.
__device__ __forceinline__ void tdm_load_b4(const float* gsrc,
                                            unsigned lds_byte_addr) {
  unsigned long long ga = (unsigned long long)(uintptr_t)gsrc;
  v4u g0;
  g0.x = 1u;                                   // count=1, user mode
  g0.y = lds_byte_addr;                        // lds_addr
  g0.z = (unsigned)ga;                         // global_addr[31:0]
  g0.w = (unsigned)((ga >> 32) & 0x01FFFFFFu)  // global_addr[56:32]
         | (2u << 30);                         // type=2 ("image")
  v8i g1;
  g1[0] = (int)((2u << 16)     // data_size = 4B
                | (1u << 20)   // pad_enable
                | (7u << 22)   // pad_interval: every 256 DWORDs
                | (1u << 25)); // pad_amount: 2 DWORDs
  g1[1] = (int)(256u << 16);   // tensor_dim0 = 256 (bits 79:48, low half)
  g1[2] = (int)(8192u << 16);  // tensor_dim0 hi=0 | tensor_dim1 lo = 8192
  g1[3] = (int)(256u << 16);   // tensor_dim1 hi=0 | tile_dim0 = 256
  g1[4] = 4;                   // tile_dim1 = 4 rows, tile_dim2 = 0
  g1[5] = 256;                 // tensor_dim0_stride = 256 elements
  g1[6] = 0;
  g1[7] = 0;
  v4i g2 = {0, 0, 0, 0};
  v4i g3 = {0, 0, 0, 0};
  v8i g4 = {0, 0, 0, 0, 0, 0, 0, 0};
  __builtin_amdgcn_tensor_load_to_lds(g0, g1, g2, g3, g4, 0);
}
#endif

// ---------------- init accumulators ----------------
__global__ void k_init(float* __restrict__ ap, float* __restrict__ counts,
                       float* __restrict__ commit_acc) {
  int i = blockIdx.x * 256 + threadIdx.x;
  if (i < K_CODES) { ap[i] = 0.f; counts[i] = 0.f; }
  if (i == 0) *commit_acc = 0.f;
}

// ---------------- row-wise L2 normalize (z and emb) ----------------
__global__ __launch_bounds__(256) void k_normalize(
    const float* __restrict__ z, const float* __restrict__ emb,
    float* __restrict__ zn, float* __restrict__ en,
    float* __restrict__ sumzz, float* __restrict__ sumee) {
  const int wave = threadIdx.x >> 5, lane = threadIdx.x & 31;
  const int row = blockIdx.x * 8 + wave;  // 0..N+K-1
  const float* src;
  float* dst;
  float* snn;
  if (row < N_ROWS) {
    src = z + (size_t)row * DIM; dst = zn + (size_t)row * DIM; snn = sumzz + row;
  } else {
    int r = row - N_ROWS;
    src = emb + (size_t)r * DIM; dst = en + (size_t)r * DIM; snn = sumee + r;
  }
  float v[8], s = 0.f;
#pragma unroll
  for (int i = 0; i < 8; ++i) { v[i] = src[lane + 32 * i]; s += v[i] * v[i]; }
#pragma unroll
  for (int off = 1; off < 32; off <<= 1) s += __shfl_xor(s, off, 32);
  float inv = 1.0f / fmaxf(sqrtf(s), 1e-12f);
#pragma unroll
  for (int i = 0; i < 8; ++i) dst[lane + 32 * i] = v[i] * inv;
  if (lane == 0) *snn = s * inv * inv;  // sum(normalized^2), ~1.0
}

// ======================= GEMM tile helpers (device) ========================
// A (zn block) kept in LDS as float2 pairs: Af2[(d/2)*64 + r] = {A[r][d], A[r][d+1]}
// B tile row-major with padded stride: Bsh[c*B_ROW_DW + d]

#define GEMM_SHARED                                                   \
  __shared__ __align__(16) v2f Af2[(DIM / 2) * BLOCK_ROWS]; /* 64KB */ \
  __shared__ __align__(16) float Bsh[2][KT * B_ROW_DW];     /* 33KB */

#define STAGE_A()                                                        \
  do {                                                                   \
    for (int i = tid; i < BLOCK_ROWS * (DIM / 2); i += 128) {            \
      int r = i >> 7, d2 = i & 127; /* coalesced global float2 reads */  \
      const v2f* zr = (const v2f*)(zn + (size_t)(row0 + r) * DIM);       \
      Af2[d2 * BLOCK_ROWS + r] = zr[d2];                                 \
    }                                                                    \
  } while (0)

#if USE_TDM
#define STAGE_B(buf, k0)                                                    \
  tdm_load_b4(en + (size_t)((k0) + wave * 4) * DIM,                         \
              (unsigned)(uintptr_t)(&Bsh[(buf)][0]) + (unsigned)wave * 4128u)
#define WAIT_B(pending) __builtin_amdgcn_s_wait_tensorcnt(pending)
#else
#define STAGE_B(buf, k0)                                           \
  do {                                                             \
    for (int i = tid; i < KT * DIM; i += 128) {                    \
      int c = i >> 8, d = i & 255;                                 \
      Bsh[(buf)][c * B_ROW_DW + d] = en[(size_t)((k0) + c) * DIM + d]; \
    }                                                              \
  } while (0)
#define WAIT_B(pending) ((void)0)
#endif

#define GEMM_TILE(acc, buf)                                              \
  do {                                                                   \
    const v2f* ag = Af2 + (size_t)h * BLOCK_ROWS + rl;                   \
    const float* bg = &Bsh[(buf)][(size_t)m * B_ROW_DW + 2 * h];         \
    _Pragma("unroll 8") for (int d2 = 0; d2 < DIM / 2; d2 += 2) {        \
      v2f a = ag[d2 * BLOCK_ROWS];                                       \
      v2f b = *(const v2f*)(bg + 2 * d2);                                \
      acc = __builtin_amdgcn_wmma_f32_16x16x4_f32(false, a, false, b,    \
                                                  (short)0, acc, false,  \
                                                  false);                \
    }                                                                    \
  } while (0)

// ---------------- pass 1: online lse + argmin over K ----------------
__global__ __launch_bounds__(128) void k_pass1(
    const float* __restrict__ zn, const float* __restrict__ en,
    const float* __restrict__ sumzz, const float* __restrict__ sumee,
    float* __restrict__ lse, int* __restrict__ idxw,
    float* __restrict__ out_idx_f) {
  GEMM_SHARED
  const int tid = threadIdx.x;
  const int wave = tid >> 5, lane = tid & 31, m = lane & 15, h = lane >> 4;
  const int row0 = blockIdx.x * BLOCK_ROWS;
  const int rl = wave * 16 + m;

  STAGE_A();
  STAGE_B(0, 0);

  float szz[8], mrun[8], srun[8], bmin[8];
  int bidx[8];
#pragma unroll
  for (int j = 0; j < 8; ++j) {
    szz[j] = sumzz[row0 + wave * 16 + j + 8 * h];
    mrun[j] = -3.0e38f; srun[j] = 0.f; bmin[j] = 3.0e38f; bidx[j] = 0;
  }

  const int NT = K_CODES / KT;
  for (int t = 0; t < NT; ++t) {
    const int cur = t & 1;
    if (t + 1 < NT) {
      STAGE_B(cur ^ 1, (t + 1) * KT);
      WAIT_B(1);  // previous tile complete; next tile may stay in flight
    } else {
      WAIT_B(0);
    }
    __syncthreads();  // tile t visible to all waves (A too, on t==0)

    v8f acc = {0.f, 0.f, 0.f, 0.f, 0.f, 0.f, 0.f, 0.f};
    GEMM_TILE(acc, cur);

    const int k0 = t * KT;
    float se = sumee[k0 + m];
#pragma unroll
    for (int j = 0; j < 8; ++j) {
      float sc = 10.0f * (szz[j] + se - 2.0f * acc[j]);  // scaled sq-distance
      if (sc < bmin[j]) { bmin[j] = sc; bidx[j] = k0 + m; }
      float tmax = grp16_max(sc);
      float nm = fmaxf(mrun[j], tmax);
      float ts = grp16_sum(__expf(sc - nm));
      srun[j] = srun[j] * __expf(mrun[j] - nm) + ts;
      mrun[j] = nm;
    }
    __syncthreads();  // done reading buffer before it is restaged
  }

#pragma unroll
  for (int j = 0; j < 8; ++j) {
    float l = mrun[j] + __logf(srun[j]);
    float bv = bmin[j]; int bi = bidx[j];
#pragma unroll
    for (int off = 1; off < 16; off <<= 1) {
      float ov = __shfl_xor(bv, off, 16);
      int oi = __shfl_xor(bi, off, 16);
      if (ov < bv || (ov == bv && oi < bi)) { bv = ov; bi = oi; }
    }
    if (m == 0) {
      int rr = row0 + wave * 16 + j + 8 * h;
      lse[rr] = l;
      idxw[rr] = bi;
      out_idx_f[rr] = (float)bi;
    }
  }
}

// ---------------- pass 2: recompute tiles, accumulate ap[k] ----------------
__global__ __launch_bounds__(128) void k_pass2(
    const float* __restrict__ zn, const float* __restrict__ en,
    const float* __restrict__ sumzz, const float* __restrict__ sumee,
    const float* __restrict__ lse, float* __restrict__ ap) {
  GEMM_SHARED
  const int tid = threadIdx.x;
  const int wave = tid >> 5, lane = tid & 31, m = lane & 15, h = lane >> 4;
  const int row0 = blockIdx.x * BLOCK_ROWS;
  const int rl = wave * 16 + m;
  const int NT = K_CODES / KT;

  STAGE_A();
  // rotate sweep start per block to spread ap[] atomic traffic over k
  const int rot = (blockIdx.x << 2) & (NT - 1);
  STAGE_B(0, rot * KT);

  float szz[8], lr[8];
#pragma unroll
  for (int j = 0; j < 8; ++j) {
    int rr = row0 + wave * 16 + j + 8 * h;
    szz[j] = sumzz[rr];
    lr[j] = lse[rr];
  }

  for (int t = 0; t < NT; ++t) {
    const int cur = t & 1;
    if (t + 1 < NT) {
      STAGE_B(cur ^ 1, (((t + 1 + rot) & (NT - 1)) * KT));
      WAIT_B(1);
    } else {
      WAIT_B(0);
    }
    __syncthreads();

    v8f acc = {0.f, 0.f, 0.f, 0.f, 0.f, 0.f, 0.f, 0.f};
    GEMM_TILE(acc, cur);

    const int k0 = ((t + rot) & (NT - 1)) * KT;
    float se = sumee[k0 + m];
    float cs = 0.f;
#pragma unroll
    for (int j = 0; j < 8; ++j) {
      float sc = 10.0f * (szz[j] + se - 2.0f * acc[j]);
      cs += __expf(sc - lr[j]);  // p[n,k]
    }
    cs += __shfl_xor(cs, 16, 32);  // add rows 8..15 half
    if (lane < 16) atomicAdd(&ap[k0 + m], cs);
    __syncthreads();
  }
}

// ---------------- gather z_q = en[idx], commit MSE, histogram ----------------
__global__ __launch_bounds__(256) void k_finalize_rows(
    const float* __restrict__ zn, const float* __restrict__ en,
    const int* __restrict__ idxw, float* __restrict__ out_zq,
    float* __restrict__ counts, float* __restrict__ commit_acc) {
  __shared__ float cpart[8];
  const int wave = threadIdx.x >> 5, lane = threadIdx.x & 31;
  const int row = blockIdx.x * 8 + wave;
  const int idx = idxw[row];
  const float* q = en + (size_t)idx * DIM;   // == l2norm(emb[idx]) == z_q_n
  const float* zr = zn + (size_t)row * DIM;  // == l2norm(z) row
  float s = 0.f;
#pragma unroll
  for (int i = 0; i < 8; ++i) {
    int d = lane + 32 * i;
    float qv = q[d];
    float df = qv - zr[d];
    s += df * df;
    out_zq[(size_t)row * DIM + d] = qv;  // straight-through forward value
  }
#pragma unroll
  for (int off = 1; off < 32; off <<= 1) s += __shfl_xor(s, off, 32);
  if (lane == 0) {
    cpart[wave] = s;
    atomicAdd(&counts[idx], 1.0f);
  }
  __syncthreads();
  if (threadIdx.x == 0) {
    float t = 0.f;
#pragma unroll
    for (int w = 0; w < 8; ++w) t += cpart[w];
    atomicAdd(commit_acc, t);
  }
}

// ---------------- final scalars: loss + perplexity ----------------
__global__ __launch_bounds__(256) void k_reduce_final(
    const float* __restrict__ ap_raw, const float* __restrict__ counts,
    const float* __restrict__ commit_acc, float* __restrict__ out_scalars) {
  __shared__ float r1[256], r2[256];
  float divs = 0.f, perps = 0.f;
  for (int k = threadIdx.x; k < K_CODES; k += 256) {
    float a = ap_raw[k] * (1.0f / (float)N_ROWS);  // mean over rows
    divs += a * __logf(a);
    float pr = counts[k] * (1.0f / (float)N_ROWS);
    perps += pr * __logf(pr + 1e-10f);
  }
  r1[threadIdx.x] = divs;
  r2[threadIdx.x] = perps;
  __syncthreads();
  for (int s = 128; s > 0; s >>= 1) {
    if (threadIdx.x < (unsigned)s) {
      r1[threadIdx.x] += r1[threadIdx.x + s];
      r2[threadIdx.x] += r2[threadIdx.x + s];
    }
    __syncthreads();
  }
  if (threadIdx.x == 0) {
    float commit = (*commit_acc) / (float)ND_TOT;
    // loss = (1+beta)*mse + diversity_loss, diversity_loss = sum(ap*log ap)
    out_scalars[0] = 1.25f * commit + r1[0];
    out_scalars[1] = __expf(-r2[0]);  // perplexity
  }
}

// ---------------------------------------------------------------------------
extern "C" void kernel_launch(void* const* d_in, const int* in_sizes, int n_in,
                              void* d_out, int out_size, void* d_ws,
                              size_t ws_size, hipStream_t stream) {
  (void)in_sizes; (void)n_in; (void)out_size; (void)ws_size;
  const float* z = (const float*)d_in[0];    // [8,1024,256] -> N x D
  const float* emb = (const float*)d_in[1];  // [K, D]

  // workspace layout (floats) ~17 MB
  float* ws = (float*)d_ws;
  float* zn = ws;
  float* en = zn + (size_t)N_ROWS * DIM;
  float* sumzz = en + (size_t)K_CODES * DIM;
  float* sumee = sumzz + N_ROWS;
  float* lse = sumee + K_CODES;
  float* ap = lse + N_ROWS;
  float* counts = ap + K_CODES;
  int* idxw = (int*)(counts + K_CODES);
  float* commit = (float*)(idxw + N_ROWS);

  float* out = (float*)d_out;
  float* out_zq = out;                         // [N*D]
  float* out_idx = out + ND_TOT;               // [N] (as float)
  float* out_scalars = out + ND_TOT + N_ROWS;  // loss, perplexity

  k_init<<<(K_CODES + 255) / 256, 256, 0, stream>>>(ap, counts, commit);
  k_normalize<<<(N_ROWS + K_CODES) / 8, 256, 0, stream>>>(z, emb, zn, en,
                                                          sumzz, sumee);
  k_pass1<<<N_ROWS / BLOCK_ROWS, 128, 0, stream>>>(zn, en, sumzz, sumee, lse,
                                                   idxw, out_idx);
  k_pass2<<<N_ROWS / BLOCK_ROWS, 128, 0, stream>>>(zn, en, sumzz, sumee, lse,
                                                   ap);
  k_finalize_rows<<<N_ROWS / 8, 256, 0, stream>>>(zn, en, idxw, out_zq, counts,
                                                  commit);
  k_reduce_final<<<1, 256, 0, stream>>>(ap, counts, commit, out_scalars);
}